// GriddingReverse_39891656245674
// MI455X (gfx1250) — compile-verified
//
#include <hip/hip_runtime.h>
#include <stdint.h>

#define EPS_F 1e-6f

// Problem shape from the reference: grid = [4, 128, 128, 128] float32.
constexpr int S  = 128;
constexpr int S2 = S * S;
constexpr int S3 = S * S * S;
constexpr int B  = 4;
constexpr int NSTORE = (3 * S) / 4;   // 96 float4 stores per output row

// One block per (b, x, y) row; threadIdx.x = z.
//  1) Async-stage the 4 input rows (x-1..x, y-1..y, 0..S-1) into LDS via
//     CDNA5 GLOBAL_LOAD_ASYNC_TO_LDS_B128 (ASYNCcnt path).
//  2) Each thread computes its cell's (px,py,pz) from LDS.
//  3) Repack through LDS so threads 0..95 emit 16B-aligned b128 stores.
__global__ __launch_bounds__(S) void gridding_reverse_kernel(
    const float* __restrict__ grid, float* __restrict__ out)
{
    __shared__ __align__(16) float rows[4][S];   // 2 KB input stage
    __shared__ __align__(16) float outs[3 * S];  // 1.5 KB output stage

    const int tid = threadIdx.x;               // z coordinate
    const int blk = blockIdx.x;                // b*S2 + x*S + y
    const int y   = blk % S;
    const int x   = (blk / S) % S;
    const int b   = blk / S2;

    // float index of this row's first output component; multiple of 4 -> 16B aligned
    const size_t rowBase =
        ((size_t)b * S3 + (size_t)x * S2 + (size_t)y * S) * 3;

    // Whole-row boundary: x==0 or y==0 -> all zeros, wide stores.
    if (x == 0 || y == 0) {
        if (tid < NSTORE) {
            float4 zv = make_float4(0.0f, 0.0f, 0.0f, 0.0f);
            *(float4*)(out + rowBase + (size_t)tid * 4) = zv;
        }
        return;
    }

    // ---- Async stage: 4 rows x 128 floats -> LDS ----
    // wave w (w = tid>>5) owns row w; lane l loads float4 #l of that row.
    {
        const int r  = tid >> 5;               // 0..3
        const int dx = r >> 1;                 // row (x-1+dx, y-1+dy)
        const int dy = r & 1;
        const float* rowPtr =
            grid + ((size_t)((b * S + (x - 1 + dx)) * S + (y - 1 + dy))) * (size_t)S;
        const unsigned gOff   = (unsigned)((tid & 31) * 16);                // bytes
        const unsigned ldsOff = (unsigned)(uintptr_t)(&rows[0][0])
                              + (unsigned)(r * S * 4)
                              + (unsigned)((tid & 31) * 16);
        asm volatile("global_load_async_to_lds_b128 %0, %1, %2"
                     :: "v"(ldsOff), "v"(gOff), "s"(rowPtr)
                     : "memory");
        asm volatile("s_wait_asynccnt 0" ::: "memory");
    }
    __syncthreads();

    // ---- Compute this z-cell ----
    const int z = tid;
    float px = 0.0f, py = 0.0f, pz = 0.0f;
    if (z != 0) {
        // rows[0]=(x-1,y-1) rows[1]=(x-1,y) rows[2]=(x,y-1) rows[3]=(x,y)
        const float a0 = rows[0][z - 1], a1 = rows[0][z];
        const float b0 = rows[1][z - 1], b1 = rows[1][z];
        const float c0 = rows[2][z - 1], c1 = rows[2][z];
        const float d0 = rows[3][z - 1], d1 = rows[3][z];

        const float x_hi = c0 + c1 + d0 + d1;   // corners with dx=1
        const float x_lo = a0 + a1 + b0 + b1;   // corners with dx=0
        const float y_hi = b0 + b1 + d0 + d1;   // dy=1
        const float y_lo = a0 + a1 + c0 + c1;   // dy=0
        const float z_hi = a1 + b1 + c1 + d1;   // dz=1
        const float z_lo = a0 + b0 + c0 + d0;   // dz=0

        const float inv   = 1.0f / (x_hi + x_lo + EPS_F);
        const float half  = (float)(S / 2);
        const float scale = 2.0f / (float)S;

        const float fx = (float)x, fy = (float)y, fz = (float)z;
        px = ((x_hi * fx + x_lo * (fx - 1.0f)) * inv - half) * scale;
        py = ((y_hi * fy + y_lo * (fy - 1.0f)) * inv - half) * scale;
        pz = ((z_hi * fz + z_lo * (fz - 1.0f)) * inv - half) * scale;
    }

    // ---- Repack through LDS for fully coalesced b128 stores ----
    outs[z * 3 + 0] = px;    // 3z mod 64 hits distinct banks (gcd(3,64)=1)
    outs[z * 3 + 1] = py;
    outs[z * 3 + 2] = pz;
    __syncthreads();

    if (tid < NSTORE) {
        float4 v = *(const float4*)&outs[tid * 4];          // 16B-aligned ds read
        *(float4*)(out + rowBase + (size_t)tid * 4) = v;    // 16B-aligned b128 store
    }
}

extern "C" void kernel_launch(void* const* d_in, const int* in_sizes, int n_in,
                              void* d_out, int out_size, void* d_ws, size_t ws_size,
                              hipStream_t stream) {
    (void)in_sizes; (void)n_in; (void)out_size; (void)d_ws; (void)ws_size;
    const float* grid = (const float*)d_in[0];
    float* out = (float*)d_out;
    const int nBlocks = B * S2;   // one block per (b, x, y) row
    gridding_reverse_kernel<<<nBlocks, S, 0, stream>>>(grid, out);
}